// GNNPEEncoder_9938554322955
// MI455X (gfx1250) — compile-verified
//
#include <hip/hip_runtime.h>

typedef float v2f __attribute__((ext_vector_type(2)));
typedef float v8f __attribute__((ext_vector_type(8)));

constexpr int N  = 65536;
constexpr int E  = 1048576;
constexpr int D  = 64;    // layer dim
constexpr int H  = 128;   // hidden
constexpr int ED = 256;   // final embedding
constexpr float BN_EPS = 1e-5f;

static __device__ __forceinline__ v8f wmma_f32(v2f a, v2f b, v8f c) {
  // D = A(16x4,f32) x B(4x16,f32) + C(16x16,f32)
  return __builtin_amdgcn_wmma_f32_16x16x4_f32(false, a, false, b, (short)0, c,
                                               false, false);
}

// ---------------------------------------------------------------- embed
__global__ __launch_bounds__(256) void k_embed(const float* __restrict__ xin,
                                               const float* __restrict__ Win,
                                               const float* __restrict__ bin,
                                               float* __restrict__ x) {
  int gid = blockIdx.x * 256 + threadIdx.x;   // N*16 threads (float4 each)
  int n = gid >> 4, c = (gid & 15) << 2;
  float xv = xin[n];
  const float4 w = *(const float4*)(Win + c);
  const float4 b = *(const float4*)(bin + c);
  float4 o;
  o.x = xv * w.x + b.x; o.y = xv * w.y + b.y;
  o.z = xv * w.z + b.z; o.w = xv * w.w + b.w;
  *(float4*)(x + (size_t)n * D + c) = o;
}

// ---------------------------------------------------------------- zero
__global__ __launch_bounds__(256) void k_zero4(float4* __restrict__ p, int n4) {
  int gid = blockIdx.x * 256 + threadIdx.x;
  if (gid < n4) p[gid] = make_float4(0.f, 0.f, 0.f, 0.f);
}

// ---------------------------------------------------------------- edge scatter
// 16 threads per edge, one float4 column-group each.
__global__ __launch_bounds__(256) void k_scatter(const int* __restrict__ ei,
                                                 const float* __restrict__ x,
                                                 float* __restrict__ agg) {
  int gid = blockIdx.x * 256 + threadIdx.x;   // E*16 threads
  int e = gid >> 4, c = (gid & 15) << 2;
  int src = ei[e];
  int dst = ei[E + e];
  const float4 v = *(const float4*)(x + (size_t)src * D + c);
  float* p = agg + (size_t)dst * D + c;
  atomicAdd(p + 0, v.x);
  atomicAdd(p + 1, v.y);
  atomicAdd(p + 2, v.z);
  atomicAdd(p + 3, v.w);
}

// ---------------------------------------------------------------- GEMM1
// T1 = ((1+eps)*x + agg) @ W1 + b1   [N,128]; accumulate col sum/sumsq.
// Block: 128 thr = 4 waves, 64 rows; wave owns 2 col-tiles (32 cols).
__global__ __launch_bounds__(128) void k_gemm1(
    const float* __restrict__ x, const float* __restrict__ agg,
    const float* __restrict__ epsp, const float* __restrict__ W1l,
    const float* __restrict__ b1l, float* __restrict__ T1,
    float* __restrict__ sum1, float* __restrict__ sq1) {
  __shared__ float hs[64 * 68];                 // padded: bank-conflict-free
  const int rowBase = blockIdx.x * 64;
  const float ep = 1.0f + epsp[0];
  for (int i = threadIdx.x; i < 64 * 16; i += 128) {
    int r = i >> 4, c = (i & 15) << 2;
    const float4 xv = *(const float4*)(x + (size_t)(rowBase + r) * D + c);
    const float4 av = *(const float4*)(agg + (size_t)(rowBase + r) * D + c);
    float* d = &hs[r * 68 + c];
    d[0] = ep * xv.x + av.x; d[1] = ep * xv.y + av.y;
    d[2] = ep * xv.z + av.z; d[3] = ep * xv.w + av.w;
  }
  __syncthreads();
  const int wave = threadIdx.x >> 5, lane = threadIdx.x & 31;
  const int ln = lane & 15, half = lane >> 4;
  const int colBase = wave * 32;
  const float bias0 = b1l[colBase + ln];
  const float bias1 = b1l[colBase + 16 + ln];
  float s0 = 0.f, q0 = 0.f, s1 = 0.f, q1 = 0.f;
  for (int rt = 0; rt < 4; ++rt) {
    v8f acc0 = {}; v8f acc1 = {};
    const float* hrow = &hs[(rt * 16 + ln) * 68];
    for (int k = 0; k < D; k += 4) {
      int kk = k + 2 * half;
      v2f a; a.x = hrow[kk]; a.y = hrow[kk + 1];
      const float* w = W1l + kk * H + colBase;
      v2f b0; b0.x = w[ln];      b0.y = w[H + ln];
      v2f b1v; b1v.x = w[16 + ln]; b1v.y = w[H + 16 + ln];
      acc0 = wmma_f32(a, b0, acc0);
      acc1 = wmma_f32(a, b1v, acc1);
    }
    for (int v = 0; v < 8; ++v) {
      float t0 = acc0[v] + bias0;
      float t1 = acc1[v] + bias1;
      int row = rowBase + rt * 16 + v + 8 * half;
      T1[(size_t)row * H + colBase + ln] = t0;
      T1[(size_t)row * H + colBase + 16 + ln] = t1;
      s0 += t0; q0 += t0 * t0; s1 += t1; q1 += t1 * t1;
    }
  }
  s0 += __shfl_xor(s0, 16, 32); q0 += __shfl_xor(q0, 16, 32);
  s1 += __shfl_xor(s1, 16, 32); q1 += __shfl_xor(q1, 16, 32);
  if (half == 0) {
    atomicAdd(&sum1[colBase + ln], s0);      atomicAdd(&sq1[colBase + ln], q0);
    atomicAdd(&sum1[colBase + 16 + ln], s1); atomicAdd(&sq1[colBase + 16 + ln], q1);
  }
}

// ---------------------------------------------------------------- BN prep
// Fold batch stats into affine: y = t*a + c
__global__ __launch_bounds__(128) void k_bnprep(
    const float* __restrict__ sum, const float* __restrict__ sq,
    const float* __restrict__ g, const float* __restrict__ be,
    float* __restrict__ a, float* __restrict__ c, int C) {
  int j = threadIdx.x;
  if (j < C) {
    const float invN = 1.0f / (float)N;
    float m = sum[j] * invN;
    float var = sq[j] * invN - m * m;
    float inv = rsqrtf(var + BN_EPS);
    float aj = g[j] * inv;
    a[j] = aj;
    c[j] = be[j] - m * aj;
  }
}

// ---------------------------------------------------------------- GEMM2
// T2 = relu(T1*a1+c1) @ W2 + b2   [N,64]; accumulate col stats.
// Block: 128 thr = 4 waves, 64 rows; wave owns 1 col-tile (16 cols), K=128.
__global__ __launch_bounds__(128) void k_gemm2(
    const float* __restrict__ T1, const float* __restrict__ a1,
    const float* __restrict__ c1, const float* __restrict__ W2l,
    const float* __restrict__ b2l, float* __restrict__ T2,
    float* __restrict__ sum2, float* __restrict__ sq2) {
  __shared__ float hs[64 * 132];
  __shared__ float aS[H], cS[H];
  if (threadIdx.x < H) { aS[threadIdx.x] = a1[threadIdx.x]; cS[threadIdx.x] = c1[threadIdx.x]; }
  __syncthreads();
  const int rowBase = blockIdx.x * 64;
  for (int i = threadIdx.x; i < 64 * 32; i += 128) {
    int r = i >> 5, c = (i & 31) << 2;
    const float4 t = *(const float4*)(T1 + (size_t)(rowBase + r) * H + c);
    float* d = &hs[r * 132 + c];
    d[0] = fmaxf(t.x * aS[c + 0] + cS[c + 0], 0.f);
    d[1] = fmaxf(t.y * aS[c + 1] + cS[c + 1], 0.f);
    d[2] = fmaxf(t.z * aS[c + 2] + cS[c + 2], 0.f);
    d[3] = fmaxf(t.w * aS[c + 3] + cS[c + 3], 0.f);
  }
  __syncthreads();
  const int wave = threadIdx.x >> 5, lane = threadIdx.x & 31;
  const int ln = lane & 15, half = lane >> 4;
  const int colBase = wave * 16;
  const float bias = b2l[colBase + ln];
  float s = 0.f, q = 0.f;
  for (int rt = 0; rt < 4; ++rt) {
    v8f acc = {};
    const float* hrow = &hs[(rt * 16 + ln) * 132];
    for (int k = 0; k < H; k += 4) {
      int kk = k + 2 * half;
      v2f a; a.x = hrow[kk]; a.y = hrow[kk + 1];
      const float* w = W2l + kk * D + colBase;
      v2f b; b.x = w[ln]; b.y = w[D + ln];
      acc = wmma_f32(a, b, acc);
    }
    for (int v = 0; v < 8; ++v) {
      float t = acc[v] + bias;
      int row = rowBase + rt * 16 + v + 8 * half;
      T2[(size_t)row * D + colBase + ln] = t;
      s += t; q += t * t;
    }
  }
  s += __shfl_xor(s, 16, 32); q += __shfl_xor(q, 16, 32);
  if (half == 0) {
    atomicAdd(&sum2[colBase + ln], s);
    atomicAdd(&sq2[colBase + ln], q);
  }
}

// ---------------------------------------------------------------- residual update
__global__ __launch_bounds__(256) void k_update(const float* __restrict__ T2,
                                                const float* __restrict__ a2,
                                                const float* __restrict__ c2,
                                                float* __restrict__ x) {
  int gid = blockIdx.x * 256 + threadIdx.x;   // N*16 threads
  int n = gid >> 4, c = (gid & 15) << 2;
  const float4 t = *(const float4*)(T2 + (size_t)n * D + c);
  float4 xv = *(const float4*)(x + (size_t)n * D + c);
  xv.x += fmaxf(t.x * a2[c + 0] + c2[c + 0], 0.f);
  xv.y += fmaxf(t.y * a2[c + 1] + c2[c + 1], 0.f);
  xv.z += fmaxf(t.z * a2[c + 2] + c2[c + 2], 0.f);
  xv.w += fmaxf(t.w * a2[c + 3] + c2[c + 3], 0.f);
  *(float4*)(x + (size_t)n * D + c) = xv;
}

// ---------------------------------------------------------------- final projection
// out = x @ Wf + bf   [N,256]. Block: 256 thr = 8 waves, 64 rows, 2 tiles/wave.
__global__ __launch_bounds__(256) void k_gemmf(const float* __restrict__ x,
                                               const float* __restrict__ Wf,
                                               const float* __restrict__ bf,
                                               float* __restrict__ out) {
  __shared__ float hs[64 * 68];
  const int rowBase = blockIdx.x * 64;
  for (int i = threadIdx.x; i < 64 * 16; i += 256) {
    int r = i >> 4, c = (i & 15) << 2;
    const float4 xv = *(const float4*)(x + (size_t)(rowBase + r) * D + c);
    float* d = &hs[r * 68 + c];
    d[0] = xv.x; d[1] = xv.y; d[2] = xv.z; d[3] = xv.w;
  }
  __syncthreads();
  const int wave = threadIdx.x >> 5, lane = threadIdx.x & 31;
  const int ln = lane & 15, half = lane >> 4;
  const int colBase = wave * 32;
  const float bias0 = bf[colBase + ln], bias1 = bf[colBase + 16 + ln];
  for (int rt = 0; rt < 4; ++rt) {
    v8f acc0 = {}; v8f acc1 = {};
    const float* hrow = &hs[(rt * 16 + ln) * 68];
    for (int k = 0; k < D; k += 4) {
      int kk = k + 2 * half;
      v2f a; a.x = hrow[kk]; a.y = hrow[kk + 1];
      const float* w = Wf + kk * ED + colBase;
      v2f b0; b0.x = w[ln];        b0.y = w[ED + ln];
      v2f b1v; b1v.x = w[16 + ln]; b1v.y = w[ED + 16 + ln];
      acc0 = wmma_f32(a, b0, acc0);
      acc1 = wmma_f32(a, b1v, acc1);
    }
    for (int v = 0; v < 8; ++v) {
      int row = rowBase + rt * 16 + v + 8 * half;
      out[(size_t)row * ED + colBase + ln] = acc0[v] + bias0;
      out[(size_t)row * ED + colBase + 16 + ln] = acc1[v] + bias1;
    }
  }
}

// ---------------------------------------------------------------- launcher
extern "C" void kernel_launch(void* const* d_in, const int* in_sizes, int n_in,
                              void* d_out, int out_size, void* d_ws, size_t ws_size,
                              hipStream_t stream) {
  (void)in_sizes; (void)n_in; (void)out_size; (void)ws_size;
  const int*   ei  = (const int*)d_in[0];
  // d_in[1] = batch: unused (batch = repeat(arange(16),4096), layout is implicit)
  const float* xin = (const float*)d_in[2];
  const float* Win = (const float*)d_in[3];
  const float* bin = (const float*)d_in[4];
  const float* eps = (const float*)d_in[5];
  const float* W1  = (const float*)d_in[6];
  const float* b1  = (const float*)d_in[7];
  const float* g1  = (const float*)d_in[8];
  const float* be1 = (const float*)d_in[9];
  const float* W2  = (const float*)d_in[10];
  const float* b2  = (const float*)d_in[11];
  const float* g2  = (const float*)d_in[12];
  const float* be2 = (const float*)d_in[13];
  const float* Wf  = (const float*)d_in[14];
  const float* bf  = (const float*)d_in[15];
  float* out = (float*)d_out;

  // workspace carve-up (~64 MB + 3 KB)
  float* ws  = (float*)d_ws;
  float* x   = ws;                         // [N,64]
  float* agg = x   + (size_t)N * D;        // [N,64]  (reused as T2)
  float* T1  = agg + (size_t)N * D;        // [N,128]
  float* st  = T1  + (size_t)N * H;        // stats: 768 floats
  float* sum1 = st;        float* sq1 = st + 128;
  float* sum2 = st + 256;  float* sq2 = st + 320;
  float* aC1  = st + 384;  float* cC1 = st + 512;
  float* aC2  = st + 640;  float* cC2 = st + 704;

  k_embed<<<N * 16 / 256, 256, 0, stream>>>(xin, Win, bin, x);

  for (int l = 0; l < 4; ++l) {
    k_zero4<<<N * 16 / 256, 256, 0, stream>>>((float4*)agg, N * 16);
    k_zero4<<<1, 256, 0, stream>>>((float4*)st, 96);           // sum/sq = 384 floats
    k_scatter<<<E * 16 / 256, 256, 0, stream>>>(ei, x, agg);
    k_gemm1<<<N / 64, 128, 0, stream>>>(x, agg, eps + l,
                                        W1 + (size_t)l * D * H, b1 + l * H,
                                        T1, sum1, sq1);
    k_bnprep<<<1, 128, 0, stream>>>(sum1, sq1, g1 + l * H, be1 + l * H, aC1, cC1, H);
    k_gemm2<<<N / 64, 128, 0, stream>>>(T1, aC1, cC1,
                                        W2 + (size_t)l * H * D, b2 + l * D,
                                        agg /* T2 */, sum2, sq2);
    k_bnprep<<<1, 128, 0, stream>>>(sum2, sq2, g2 + l * D, be2 + l * D, aC2, cC2, D);
    k_update<<<N * 16 / 256, 256, 0, stream>>>(agg, aC2, cC2, x);
  }

  k_gemmf<<<N / 64, 256, 0, stream>>>(x, Wf, bf, out);
}